// FNO1d_32418413150515
// MI455X (gfx1250) — compile-verified
//
#include <hip/hip_runtime.h>
#include <math.h>
#include <stdint.h>

// FNO1d on MI455X (gfx1250, wave32).
// Bandwidth-bound (~2.5-3 GB traffic vs ~40 GF compute at 23.3 TB/s) -> full f32,
// V_WMMA_F32_16X16X4_F32 for all channel-mixing GEMMs, async global->LDS staging,
// one-shot weight transpose to kill the 16x strided-gather amplification.

#define BATCH 64
#define SLEN  8192
#define WID   64
#define MODES 256
#define NSUB  32          // 8192 = 32 * 256 decimation
#define SEQ_PITCH 257     // LDS pitch to avoid bank conflicts

typedef __attribute__((ext_vector_type(2))) float v2f;
typedef __attribute__((ext_vector_type(4))) float v4f;
typedef __attribute__((ext_vector_type(8))) float v8f;

__device__ __forceinline__ float gelu_exact(float x) {
  return 0.5f * x * (1.0f + erff(x * 0.7071067811865476f));
}

__device__ __forceinline__ int bitrev8(int k) {
  return (int)(__brev((unsigned)k) >> 24);
}

// D = A(16x4,f32) * B(4x16,f32) + C(16x16,f32), full-precision WMMA.
__device__ __forceinline__ v8f wmma_f32(v2f a, v2f b, v8f c) {
  return __builtin_amdgcn_wmma_f32_16x16x4_f32(false, a, false, b, (short)0, c,
                                               false, false);
}

// Async global->LDS copy (GV mode: 64-bit per-lane address, saddr = off).
// Tracked by ASYNCcnt; per-lane LDS destination supports arbitrary swizzles.
__device__ __forceinline__ void async_g2l_b32(unsigned lds_off, const float* gptr) {
  asm volatile("global_load_async_to_lds_b32 %0, %1, off"
               :: "v"(lds_off), "v"((unsigned long long)(uintptr_t)gptr)
               : "memory");
}
__device__ __forceinline__ void wait_async0() {
  asm volatile("s_wait_asynccnt 0x0" ::: "memory");
}
__device__ __forceinline__ unsigned lds_off32(const void* p) {
  return (unsigned)(uintptr_t)p;   // low 32 bits of generic addr = LDS offset
}

// ---------------------------------------------------------------- twiddles
// tw[2t] = cos(2*pi*t/8192), tw[2t+1] = sin(2*pi*t/8192)
__global__ void k_twiddle(float* __restrict__ tw) {
  int t = blockIdx.x * 256 + threadIdx.x;
  float th = 6.283185307179586f * (float)t * (1.0f / 8192.0f);
  tw[2 * t]     = cosf(th);
  tw[2 * t + 1] = sinf(th);
}

// --------------------------------------------------------- weight transpose
// sw[l][i][o][k] (k contiguous) -> wt[l][k][i*64+o].  Reads coalesced (v4f
// along k); 4B scattered writes land in the L2-resident 33.5MB destination.
__global__ void k_wt(const float* __restrict__ swr, const float* __restrict__ swi,
                     float* __restrict__ wtr, float* __restrict__ wti) {
  int idx = blockIdx.x * 256 + threadIdx.x;   // < 4*4096*64 = 2^20
  int kq = idx & 63;
  int io = (idx >> 6) & 4095;
  int l  = idx >> 18;
  size_t src = ((size_t)(l * 4096 + io)) * 256 + (size_t)kq * 4;
  v4f ar = *(const v4f*)(swr + src);
  v4f ai = *(const v4f*)(swi + src);
  size_t dst = (size_t)l * (MODES * 4096) + io;
  for (int j = 0; j < 4; ++j) {
    wtr[dst + (size_t)(kq * 4 + j) * 4096] = ar[j];
    wti[dst + (size_t)(kq * 4 + j) * 4096] = ai[j];
  }
}

// ---------------------------------------------------------------- fc0 lift
__global__ void k_fc0(const float* __restrict__ u, const float* __restrict__ w,
                      const float* __restrict__ bias, float* __restrict__ x) {
  int idx = blockIdx.x * 256 + threadIdx.x;          // < 2^25
  int s = idx & (SLEN - 1);
  int c = (idx >> 13) & 63;
  int b = idx >> 19;
  float g = (float)s * (1.0f / (float)(SLEN - 1));
  x[idx] = u[b * SLEN + s] * w[c] + g * w[WID + c] + bias[c];
}

// -------------------------------------------------- 32 x 256-pt radix-2 FFT
// DIF (Gentleman-Sande), natural input -> bit-reversed output.
template <int SIGNPOS>
__device__ __forceinline__ void fft256x32(float* sre, float* sim,
                                          const float* __restrict__ tw, int tid) {
  int hs = 7;
  for (int len = 256; len >= 2; len >>= 1, --hs) {
    int half = len >> 1;
    for (int it = 0; it < 16; ++it) {
      int bf  = tid + 256 * it;      // 4096 butterflies / stage
      int seq = bf >> 7;
      int j   = bf & 127;
      int grp = j >> hs;
      int pos = j & (half - 1);
      int i0  = seq * SEQ_PITCH + grp * len + pos;
      int i1  = i0 + half;
      float ar = sre[i0], ai = sim[i0];
      float br = sre[i1], bi = sim[i1];
      sre[i0] = ar + br; sim[i0] = ai + bi;
      float dr = ar - br, di = ai - bi;
      int ti = pos << (12 - hs);     // pos * (8192/len)
      float c  = tw[2 * ti];
      float sn = SIGNPOS ? tw[2 * ti + 1] : -tw[2 * ti + 1];
      sre[i1] = dr * c - di * sn;
      sim[i1] = dr * sn + di * c;
    }
    __syncthreads();
  }
}

// ------------------------------------------------------------- forward FFT
// One workgroup per (b,c) row: X[k], k<256 via 32 sub-FFTs + twiddle combine.
// Row load uses async-to-LDS with the decimation swizzle applied in the LDS
// destination address.  Output layout xf[k][b*64+i] (coalesced GEMM staging).
__global__ void k_fft_fwd(const float* __restrict__ x, const float* __restrict__ tw,
                          float* __restrict__ xfr, float* __restrict__ xfi) {
  __shared__ float sre[NSUB * SEQ_PITCH];
  __shared__ float sim[NSUB * SEQ_PITCH];
  const int tid = threadIdx.x;
  const int row = blockIdx.x;                    // b*64 + c
  const float* xr = x + (size_t)row * SLEN;
  for (int it = 0; it < 32; ++it) {              // seq[r][m] = x[32m+r]
    int n = tid + 256 * it;
    int r = n & 31, m = n >> 5;
    async_g2l_b32(lds_off32(&sre[r * SEQ_PITCH + m]), xr + n);
    sim[r * SEQ_PITCH + m] = 0.0f;
  }
  wait_async0();
  __syncthreads();
  fft256x32<0>(sre, sim, tw, tid);
  // X[k] = sum_r e^{-2pi i k r/8192} * S_r[k]   (S_r at bit-reversed index)
  int k = tid;
  int kbr = bitrev8(k);
  float accr = 0.0f, acci = 0.0f;
  for (int r = 0; r < 32; ++r) {
    int ti = (k * r) & (SLEN - 1);
    float c = tw[2 * ti], sn = -tw[2 * ti + 1];
    float vr = sre[r * SEQ_PITCH + kbr];
    float vi = sim[r * SEQ_PITCH + kbr];
    accr += vr * c - vi * sn;
    acci += vr * sn + vi * c;
  }
  xfr[(size_t)k * (BATCH * WID) + row] = accr;   // scattered store -> L2 (8.4MB hot)
  xfi[(size_t)k * (BATCH * WID) + row] = acci;
}

// ----------------------------------------------------- per-mode complex GEMM
// Om[b*64+o][k] = sum_i Xf[k][b*64+i] * (Wr+iWi)[k][i][o].  One block per mode.
// All staging reads coalesced + async; f32 WMMA with 4 partial accumulators
// (A/B NEG is illegal for f32 operands).
__global__ void k_mode_gemm(const float* __restrict__ xfr, const float* __restrict__ xfi,
                            const float* __restrict__ wr, const float* __restrict__ wi,
                            float* __restrict__ omr, float* __restrict__ omi) {
  __shared__ float Are[64 * 65], Aim[64 * 65];   // [b][i], padded
  __shared__ float Bre[64 * 64], Bim[64 * 64];   // [i][o]
  const int tid = threadIdx.x;
  const int k = blockIdx.x;
  const size_t k4096 = (size_t)k * 4096;
  for (int t = tid; t < 4096; t += 256) {
    int bb = t >> 6, ii = t & 63;
    async_g2l_b32(lds_off32(&Are[bb * 65 + ii]), xfr + k4096 + t);
    async_g2l_b32(lds_off32(&Aim[bb * 65 + ii]), xfi + k4096 + t);
    async_g2l_b32(lds_off32(&Bre[t]), wr + k4096 + t);
    async_g2l_b32(lds_off32(&Bim[t]), wi + k4096 + t);
  }
  wait_async0();
  __syncthreads();
  int wid = tid >> 5, lane = tid & 31;
  int lrow = lane & 15, lk = (lane >> 4) * 2;
  for (int tt = 0; tt < 2; ++tt) {
    int tile = wid * 2 + tt;                     // 16 tiles = 4x4 of 16x16
    int m0 = (tile >> 2) * 16, n0 = (tile & 3) * 16;
    v8f prr = {}, pii = {}, pri = {}, pir = {};
    for (int kc = 0; kc < 64; kc += 4) {
      v2f ar, ai, br, bi;
      const float* Ar0 = &Are[(m0 + lrow) * 65 + kc + lk];
      const float* Ai0 = &Aim[(m0 + lrow) * 65 + kc + lk];
      ar.x = Ar0[0]; ar.y = Ar0[1];
      ai.x = Ai0[0]; ai.y = Ai0[1];
      br.x = Bre[(kc + lk) * 64 + n0 + lrow];
      br.y = Bre[(kc + lk + 1) * 64 + n0 + lrow];
      bi.x = Bim[(kc + lk) * 64 + n0 + lrow];
      bi.y = Bim[(kc + lk + 1) * 64 + n0 + lrow];
      prr = wmma_f32(ar, br, prr);
      pii = wmma_f32(ai, bi, pii);
      pri = wmma_f32(ar, bi, pri);
      pir = wmma_f32(ai, br, pir);
    }
    int col = n0 + lrow;
    int rbase = m0 + (lane >> 4) * 8;
    for (int i = 0; i < 8; ++i) {
      size_t o = ((size_t)(rbase + i) * 64 + col) * MODES + k;  // Om[row][k]
      omr[o] = prr[i] - pii[i];                  // Re = ArBr - AiBi
      omi[o] = pri[i] + pir[i];                  // Im = ArBi + AiBr
    }
  }
}

// ----------------------------------------------------------- pointwise conv
// y[b][o][s] = sum_i W[o][i] x[b][i][s] + bias[o].  M=64(o), K=64(i), N=s.
__global__ void k_pw(const float* __restrict__ x, const float* __restrict__ wgt,
                     const float* __restrict__ bias, float* __restrict__ y) {
  const int tid = threadIdx.x;
  const int b  = blockIdx.x >> 6;
  const int sc = (blockIdx.x & 63) * 128;        // 128 s-cols per block
  int wid = tid >> 5, lane = tid & 31;
  int lrow = lane & 15, lk = (lane >> 4) * 2;
  const float* xb = x + (size_t)b * (WID * SLEN);
  int col = sc + wid * 16 + lrow;
  __builtin_prefetch(&xb[col], 0, 3);
  v8f zero = {};
  v8f acc[4];
  for (int i = 0; i < 4; ++i) acc[i] = zero;
  for (int kc = 0; kc < 64; kc += 4) {
    v2f bf;
    bf.x = xb[(size_t)(kc + lk) * SLEN + col];
    bf.y = xb[(size_t)(kc + lk + 1) * SLEN + col];
    for (int mt = 0; mt < 4; ++mt) {
      v2f af;
      const float* wrow = &wgt[(mt * 16 + lrow) * 64 + kc + lk];
      af.x = wrow[0]; af.y = wrow[1];
      acc[mt] = wmma_f32(af, bf, acc[mt]);
    }
  }
  int rb = (lane >> 4) * 8;
  for (int mt = 0; mt < 4; ++mt)
    for (int i = 0; i < 8; ++i) {
      int row = mt * 16 + rb + i;
      y[(size_t)b * (WID * SLEN) + (size_t)row * SLEN + col] = acc[mt][i] + bias[row];
    }
}

// -------------------------------------------- inverse FFT + add + (gelu)
// x[32m+r] = (2/S) Re( FFT+_{256}( eps_k X[k] e^{+2pi i k r/8192} )[m] )
// Fused: xio = maybe_gelu(xio + x1).  One block per (b,o) row; Om read coalesced.
__global__ void k_ifft(const float* __restrict__ omr, const float* __restrict__ omi,
                       const float* __restrict__ tw, float* __restrict__ xio,
                       int do_act) {
  __shared__ float sre[NSUB * SEQ_PITCH];
  __shared__ float sim[NSUB * SEQ_PITCH];
  const int tid = threadIdx.x;
  const int row = blockIdx.x;                    // b*64 + o
  {
    int k = tid;
    float Xr = omr[(size_t)row * MODES + k];
    float Xi = omi[(size_t)row * MODES + k];
    if (k == 0) { Xr *= 0.5f; Xi *= 0.5f; }      // eps_0 = 1/2
    for (int r = 0; r < 32; ++r) {
      int ti = (k * r) & (SLEN - 1);
      float c = tw[2 * ti], sn = tw[2 * ti + 1]; // e^{+i}
      sre[r * SEQ_PITCH + k] = Xr * c - Xi * sn;
      sim[r * SEQ_PITCH + k] = Xr * sn + Xi * c;
    }
  }
  __syncthreads();
  fft256x32<1>(sre, sim, tw, tid);
  const float scale = 2.0f / (float)SLEN;
  float* xr = xio + (size_t)row * SLEN;
  for (int it = 0; it < 32; ++it) {
    int n = tid + 256 * it;
    int r = n & 31, m = n >> 5;
    float v = xr[n] + sre[r * SEQ_PITCH + bitrev8(m)] * scale;
    if (do_act) v = gelu_exact(v);
    xr[n] = v;
  }
}

// ------------------------------------------------------------- head
// out[b][s] = gelu(x^T fc1 + b1) @ fc2 + b2.  M=128 s-rows per block.
__global__ void k_head(const float* __restrict__ x, const float* __restrict__ w1,
                       const float* __restrict__ b1, const float* __restrict__ w2,
                       const float* __restrict__ b2, float* __restrict__ out) {
  __shared__ float H[128 * 130];                 // gelu(hidden), padded
  const int tid = threadIdx.x;
  const int b  = blockIdx.x >> 6;
  const int s0 = (blockIdx.x & 63) * 128;
  int wid = tid >> 5, lane = tid & 31;
  int lrow = lane & 15, lk = (lane >> 4) * 2;
  const float* xb = x + (size_t)b * (WID * SLEN) + s0;
  int mt = wid;                                  // wave owns one 16-row tile
  v8f zero = {};
  v8f acc[8];
  for (int i = 0; i < 8; ++i) acc[i] = zero;
  for (int kc = 0; kc < 64; kc += 4) {
    v2f af;                                      // A[srow][c] = x[b][c][s0+srow]
    af.x = xb[(size_t)(kc + lk) * SLEN + mt * 16 + lrow];
    af.y = xb[(size_t)(kc + lk + 1) * SLEN + mt * 16 + lrow];
    for (int nt = 0; nt < 8; ++nt) {
      v2f bf;
      bf.x = w1[(kc + lk) * 128 + nt * 16 + lrow];
      bf.y = w1[(kc + lk + 1) * 128 + nt * 16 + lrow];
      acc[nt] = wmma_f32(af, bf, acc[nt]);
    }
  }
  int rb = (lane >> 4) * 8;
  for (int nt = 0; nt < 8; ++nt) {
    int j = nt * 16 + lrow;
    float bj = b1[j];
    for (int i = 0; i < 8; ++i)
      H[(mt * 16 + rb + i) * 130 + j] = gelu_exact(acc[nt][i] + bj);
  }
  __syncthreads();
  if (tid < 128) {
    float a = b2[0];
    for (int j = 0; j < 128; ++j) a += H[tid * 130 + j] * w2[j];
    out[(size_t)b * SLEN + s0 + tid] = a;
  }
}

// ------------------------------------------------------------------ launch
extern "C" void kernel_launch(void* const* d_in, const int* in_sizes, int n_in,
                              void* d_out, int out_size, void* d_ws, size_t ws_size,
                              hipStream_t stream) {
  const float* u    = (const float*)d_in[0];
  const float* fc0w = (const float*)d_in[1];
  const float* fc0b = (const float*)d_in[2];
  const float* swr  = (const float*)d_in[3];
  const float* swi  = (const float*)d_in[4];
  const float* pww  = (const float*)d_in[5];
  const float* pwb  = (const float*)d_in[6];
  const float* fc1w = (const float*)d_in[7];
  const float* fc1b = (const float*)d_in[8];
  const float* fc2w = (const float*)d_in[9];
  const float* fc2b = (const float*)d_in[10];
  float* out = (float*)d_out;

  const size_t XBUF = (size_t)BATCH * WID * SLEN;       // 33.5M floats
  const size_t FBUF = (size_t)MODES * BATCH * WID;      // 1.05M floats
  const size_t WBUF = (size_t)4 * MODES * WID * WID;    // 4.19M floats
  float* ws  = (float*)d_ws;
  float* tw  = ws;                  // 16384 floats
  float* xA  = tw + 16384;
  float* xB  = xA + XBUF;
  float* xfr = xB + XBUF;
  float* xfi = xfr + FBUF;
  float* omr = xfi + FBUF;
  float* omi = omr + FBUF;
  float* wtr = omi + FBUF;
  float* wti = wtr + WBUF;

  k_twiddle<<<SLEN / 256, 256, 0, stream>>>(tw);
  k_wt<<<(4 * 4096 * 64) / 256, 256, 0, stream>>>(swr, swi, wtr, wti);
  k_fc0<<<(int)(XBUF / 256), 256, 0, stream>>>(u, fc0w, fc0b, xA);

  float* cur = xA;
  float* nxt = xB;
  for (int l = 0; l < 4; ++l) {
    k_fft_fwd<<<BATCH * WID, 256, 0, stream>>>(cur, tw, xfr, xfi);
    k_mode_gemm<<<MODES, 256, 0, stream>>>(xfr, xfi,
                                           wtr + (size_t)l * MODES * 4096,
                                           wti + (size_t)l * MODES * 4096,
                                           omr, omi);
    k_pw<<<BATCH * (SLEN / 128), 256, 0, stream>>>(cur, pww + l * WID * WID,
                                                   pwb + l * WID, nxt);
    k_ifft<<<BATCH * WID, 256, 0, stream>>>(omr, omi, tw, nxt, (l < 3) ? 1 : 0);
    float* t = cur; cur = nxt; nxt = t;
  }
  k_head<<<BATCH * (SLEN / 128), 256, 0, stream>>>(cur, fc1w, fc1b, fc2w, fc2b, out);
}